// LightweightSupplyGNN_81853486727232
// MI455X (gfx1250) — compile-verified
//
#include <hip/hip_runtime.h>
#include <hip/hip_bf16.h>
#include <math.h>

// GCN pipeline for MI455X (gfx1250):
//   deg/rsqrt -> WMMA-f32 GEMM (epilogue-scaled by dinv) -> float4 atomic
//   edge scatter -> self-loop+bias+ReLU (x2 layers) -> fused 3-head MLP.
// GEMMs use V_WMMA_F32_16X16X4_F32 (exact fp32). One wave owns a 16x64 row
// block with 4 independent accumulators so consecutive WMMAs have no D->C
// RAW chain. W is staged in LDS (stride 80 => conflict-free half-wave rows).
// Scatter is the memory-bound hot loop; xs/agg (25.6MB each) stay in the
// 192MB L2, so per-element global_atomic_add_f32 at b128-gather granularity
// is the right path.

typedef float v2f __attribute__((ext_vector_type(2)));
typedef float v8f __attribute__((ext_vector_type(8)));

#define D_IN 128
#define H    64
#define HH   32
#define BSTRIDE 80   // LDS row stride for staged B (bank-conflict-free)

__device__ __forceinline__ float atomAddF(float* p, float v) {
  return unsafeAtomicAdd(p, v);   // hardware global_atomic_add_f32
}

__global__ void zero_f32(float* __restrict__ p, int n) {
  int i = blockIdx.x * blockDim.x + threadIdx.x;
  if (i < n) p[i] = 0.0f;
}

__global__ void degree_kernel(const int* __restrict__ dst, float* __restrict__ deg, int E) {
  int e = blockIdx.x * blockDim.x + threadIdx.x;
  if (e < E) atomAddF(&deg[dst[e]], 1.0f);
}

__global__ void dinv_kernel(float* __restrict__ dinv, int n) {
  int i = blockIdx.x * blockDim.x + threadIdx.x;
  if (i < n) dinv[i] = rsqrtf(dinv[i] + 1.0f);  // +1 for implicit self-loop
}

// C[M,64] = (A[M,K] @ B[K,64]) * rowScale[row], fp32 WMMA 16x16x4.
// One wave per 16-row block, 4 accumulators cover N=64.
// A-frag (ISA 7.12.2, 32-bit A 16x4): lane&15 -> M row, (lane>>4)*2 -> K pair.
// C/D: VGPR r -> M = r + (lane>>4)*8, N = lane&15 within tile.
__global__ __launch_bounds__(256) void gemm_wmma_rows(
    const float* __restrict__ A, const float* __restrict__ B,
    const float* __restrict__ rowScale, float* __restrict__ C,
    int M, int K) {
  __shared__ float sB[D_IN * BSTRIDE];   // up to 128x64 staged, 40KB

  for (int t = threadIdx.x; t < K * H; t += blockDim.x) {
    int row = t >> 6, col = t & 63;
    sB[row * BSTRIDE + col] = B[t];
  }
  __syncthreads();

  int rb = blockIdx.x * (blockDim.x >> 5) + (threadIdx.x >> 5);  // 16-row block
  if (rb >= (M >> 4)) return;            // wave-uniform; EXEC stays all-1s

  int lane = threadIdx.x & 31;
  int l15  = lane & 15;
  int half = lane >> 4;                  // 0 or 1
  int kb   = half * 2;                   // K sub-offset {0,2}

  const float* Arow = A + (size_t)(rb * 16 + l15) * K;

  v8f acc[4];
#pragma unroll
  for (int nt = 0; nt < 4; ++nt) acc[nt] = (v8f){0.f,0.f,0.f,0.f,0.f,0.f,0.f,0.f};

  for (int k = 0; k < K; k += 4) {
    float2 av = *(const float2*)(Arow + k + kb);   // 8B-aligned (k+kb even)
    v2f a; a.x = av.x; a.y = av.y;
    const float* b0 = &sB[(k + kb) * BSTRIDE + l15];
#pragma unroll
    for (int nt = 0; nt < 4; ++nt) {
      v2f b;
      b.x = b0[nt * 16];
      b.y = b0[nt * 16 + BSTRIDE];
      acc[nt] = __builtin_amdgcn_wmma_f32_16x16x4_f32(
          false, a, false, b, (short)0, acc[nt], false, false);
    }
  }

  int mbase = rb * 16 + half * 8;
  float* Crow = C + (size_t)mbase * H + l15;
#pragma unroll
  for (int r = 0; r < 8; ++r) {
    float sc = rowScale[mbase + r];
#pragma unroll
    for (int nt = 0; nt < 4; ++nt)
      Crow[(size_t)r * H + nt * 16] = acc[nt][r] * sc;
  }
}

// 16 threads per edge, float4 per thread:
// agg[dst, 4q..4q+3] += xs[src, 4q..4q+3] * dinv[dst]   (xs pre-scaled by dinv[src])
__global__ void scatter_kernel(const float* __restrict__ xs,
                               const int* __restrict__ src,
                               const int* __restrict__ dst,
                               const float* __restrict__ dinv,
                               float* __restrict__ agg, int E) {
  int idx = blockIdx.x * blockDim.x + threadIdx.x;
  int e = idx >> 4;
  if (e >= E) return;
  int q = idx & 15;
  int s = src[e];
  int d = dst[e];
  float nd = dinv[d];
  float4 xv = *(const float4*)(xs + (size_t)s * H + q * 4);
  float* p = agg + (size_t)d * H + q * 4;
  atomAddF(p + 0, xv.x * nd);
  atomAddF(p + 1, xv.y * nd);
  atomAddF(p + 2, xv.z * nd);
  atomAddF(p + 3, xv.w * nd);
}

// out = relu(agg + xs * dinv + bias)   (xs already carries one dinv factor)
__global__ void combine_relu(const float* __restrict__ agg,
                             const float* __restrict__ xs,
                             const float* __restrict__ dinv,
                             const float* __restrict__ bias,
                             float* __restrict__ out, int nNodes) {
  int idx = blockIdx.x * blockDim.x + threadIdx.x;
  if (idx >= nNodes * H) return;
  int i = idx >> 6;
  int j = idx & 63;
  float v = agg[idx] + xs[idx] * dinv[i] + bias[j];
  out[idx] = fmaxf(v, 0.0f);
}

// Fused 3 heads: out = sigmoid(relu(h@W1 + b1) @ W2 + b2). Weights in LDS,
// h row cached in registers (16 x float4), one node per thread.
__global__ __launch_bounds__(256) void heads_kernel(
    const float* __restrict__ h,
    const float* __restrict__ Wd1, const float* __restrict__ bd1,
    const float* __restrict__ Wd2, const float* __restrict__ bd2,
    const float* __restrict__ Wi1, const float* __restrict__ bi1,
    const float* __restrict__ Wi2, const float* __restrict__ bi2,
    const float* __restrict__ Wc1, const float* __restrict__ bc1,
    const float* __restrict__ Wc2, const float* __restrict__ bc2,
    float* __restrict__ outD, float* __restrict__ outI, float* __restrict__ outC,
    int nNodes) {
  __shared__ float sW1[3][H * HH];
  __shared__ float sB1[3][HH];
  __shared__ float sW2[3][HH];
  __shared__ float sB2[3];

  for (int t = threadIdx.x; t < H * HH; t += blockDim.x) {
    sW1[0][t] = Wd1[t];
    sW1[1][t] = Wi1[t];
    sW1[2][t] = Wc1[t];
  }
  for (int t = threadIdx.x; t < HH; t += blockDim.x) {
    sB1[0][t] = bd1[t]; sB1[1][t] = bi1[t]; sB1[2][t] = bc1[t];
    sW2[0][t] = Wd2[t]; sW2[1][t] = Wi2[t]; sW2[2][t] = Wc2[t];
  }
  if (threadIdx.x == 0) { sB2[0] = bd2[0]; sB2[1] = bi2[0]; sB2[2] = bc2[0]; }
  __syncthreads();

  int i = blockIdx.x * blockDim.x + threadIdx.x;
  if (i >= nNodes) return;

  float4 hv[16];
  const float4* hp = (const float4*)(h + (size_t)i * H);
#pragma unroll
  for (int q = 0; q < 16; ++q) hv[q] = hp[q];

#pragma unroll
  for (int hd = 0; hd < 3; ++hd) {
    float acc2 = sB2[hd];
    for (int j = 0; j < HH; ++j) {
      float s = sB1[hd][j];
#pragma unroll
      for (int q = 0; q < 16; ++q) {
        const float* w = &sW1[hd][(q * 4) * HH + j];
        s += hv[q].x * w[0 * HH] + hv[q].y * w[1 * HH]
           + hv[q].z * w[2 * HH] + hv[q].w * w[3 * HH];
      }
      s = fmaxf(s, 0.0f);
      acc2 += s * sW2[hd][j];
    }
    float r = 1.0f / (1.0f + expf(-acc2));
    if (hd == 0) outD[i] = r;
    else if (hd == 1) outI[i] = r;
    else outC[i] = r;
  }
}

extern "C" void kernel_launch(void* const* d_in, const int* in_sizes, int n_in,
                              void* d_out, int out_size, void* d_ws, size_t ws_size,
                              hipStream_t stream) {
  const float* x   = (const float*)d_in[0];
  const int*   ei  = (const int*)d_in[1];
  const float* W1  = (const float*)d_in[2];
  const float* b1  = (const float*)d_in[3];
  const float* W2  = (const float*)d_in[4];
  const float* b2  = (const float*)d_in[5];
  const float* Wd1 = (const float*)d_in[6];  const float* bd1 = (const float*)d_in[7];
  const float* Wd2 = (const float*)d_in[8];  const float* bd2 = (const float*)d_in[9];
  const float* Wi1 = (const float*)d_in[10]; const float* bi1 = (const float*)d_in[11];
  const float* Wi2 = (const float*)d_in[12]; const float* bi2 = (const float*)d_in[13];
  const float* Wc1 = (const float*)d_in[14]; const float* bc1 = (const float*)d_in[15];
  const float* Wc2 = (const float*)d_in[16]; const float* bc2 = (const float*)d_in[17];

  const int N = in_sizes[0] / D_IN;   // 100000
  const int E = in_sizes[1] / 2;      // 3200000
  const int* src = ei;
  const int* dst = ei + E;

  float* ws   = (float*)d_ws;
  float* dinv = ws;                                        // N floats
  float* xs   = ws + (((size_t)N + 511) & ~(size_t)511);   // N*H (dinv-scaled xw)
  float* h1   = xs + (size_t)N * H;                        // N*H

  float* outH = (float*)d_out;                             // N*H
  float* outD = outH + (size_t)N * H;
  float* outI = outD + N;
  float* outC = outI + N;

  const int TB = 256;
  const int NH = N * H;

  // --- symmetric-norm degree ---
  zero_f32<<<(N + TB - 1) / TB, TB, 0, stream>>>(dinv, N);
  degree_kernel<<<(E + TB - 1) / TB, TB, 0, stream>>>(dst, dinv, E);
  dinv_kernel<<<(N + TB - 1) / TB, TB, 0, stream>>>(dinv, N);

  int rowBlocks  = N / 16;                     // 6250
  int gemmBlocks = (rowBlocks + 7) / 8;        // 8 waves / 256-thread block
  int scatterBlocks = (int)(((long long)E * 16 + TB - 1) / TB);

  // --- layer 1 ---
  gemm_wmma_rows<<<gemmBlocks, 256, 0, stream>>>(x, W1, dinv, xs, N, D_IN);
  zero_f32<<<(NH + TB - 1) / TB, TB, 0, stream>>>(h1, NH);
  scatter_kernel<<<scatterBlocks, TB, 0, stream>>>(xs, src, dst, dinv, h1, E);
  combine_relu<<<(NH + TB - 1) / TB, TB, 0, stream>>>(h1, xs, dinv, b1, h1, N);

  // --- layer 2 (agg accumulates straight into d_out's h region) ---
  gemm_wmma_rows<<<gemmBlocks, 256, 0, stream>>>(h1, W2, dinv, xs, N, H);
  zero_f32<<<(NH + TB - 1) / TB, TB, 0, stream>>>(outH, NH);
  scatter_kernel<<<scatterBlocks, TB, 0, stream>>>(xs, src, dst, dinv, outH, E);
  combine_relu<<<(NH + TB - 1) / TB, TB, 0, stream>>>(outH, xs, dinv, b2, outH, N);

  // --- fused heads ---
  heads_kernel<<<(N + TB - 1) / TB, TB, 0, stream>>>(
      outH, Wd1, bd1, Wd2, bd2, Wi1, bi1, Wi2, bi2, Wc1, bc1, Wc2, bc2,
      outD, outI, outC, N);
}